// KalmanFilteredLoss_77524159693313
// MI455X (gfx1250) — compile-verified
//
#include <hip/hip_runtime.h>
#include <hip/hip_bf16.h>

#define EPS_F      1e-7f
#define GIOU_THR   0.2f

typedef __attribute__((ext_vector_type(2))) float v2f;
typedef __attribute__((ext_vector_type(8))) float v8f;

__device__ __forceinline__ float4 ld4(const float* p) { return *(const float4*)p; }

// cxcywh (x=cx,y=cy,z=w,w=h) -> xyxy
__device__ __forceinline__ float4 to_xyxy(float4 b) {
    return make_float4(b.x - 0.5f * b.z, b.y - 0.5f * b.w,
                       b.x + 0.5f * b.z, b.y + 0.5f * b.w);
}

// filtered = (prev * scale) / scale, scale = (W,H,W,H)
__device__ __forceinline__ float4 filt_adjust(float4 pb, float W, float H) {
    return make_float4((pb.x * W) / W, (pb.y * H) / H,
                       (pb.z * W) / W, (pb.w * H) / H);
}

__device__ __forceinline__ float giou_loss(float4 A, float4 Bx) {  // xyxy
    float area_a = (A.z - A.x) * (A.w - A.y);
    float area_b = (Bx.z - Bx.x) * (Bx.w - Bx.y);
    float iw = fmaxf(fminf(A.z, Bx.z) - fmaxf(A.x, Bx.x), 0.f);
    float ih = fmaxf(fminf(A.w, Bx.w) - fmaxf(A.y, Bx.y), 0.f);
    float inter = iw * ih;
    float uni = area_a + area_b - inter;
    float iou = inter / (uni + EPS_F);
    float cw = fmaxf(fmaxf(A.z, Bx.z) - fminf(A.x, Bx.x), 0.f);
    float ch = fmaxf(fmaxf(A.w, Bx.w) - fminf(A.y, Bx.y), 0.f);
    float area_c = cw * ch;
    float giou = iou - (area_c - uni) / (area_c + EPS_F);
    return 1.f - giou;
}

__device__ __forceinline__ float ciou_loss(float4 P, float4 T) {  // xyxy pred, target
    float iw = fmaxf(fminf(P.z, T.z) - fmaxf(P.x, T.x), 0.f);
    float ih = fmaxf(fminf(P.w, T.w) - fmaxf(P.y, T.y), 0.f);
    float inter = iw * ih;
    float uni = (P.z - P.x) * (P.w - P.y) + (T.z - T.x) * (T.w - T.y) - inter;
    float iou = inter / (uni + EPS_F);
    float cw = fmaxf(P.z, T.z) - fminf(P.x, T.x);
    float ch = fmaxf(P.w, T.w) - fminf(P.y, T.y);
    float c2 = cw * cw + ch * ch + EPS_F;
    float dx = (P.x + P.z) - (T.x + T.z);
    float dy = (P.y + P.w) - (T.y + T.w);
    float d2 = (dx * dx + dy * dy) * 0.25f;
    const float VK = 0.40528473456935108577f;  // 4/pi^2
    float at = atanf((T.z - T.x) / (T.w - T.y)) - atanf((P.z - P.x) / (P.w - P.y));
    float v = VK * at * at;
    float alpha = v / (1.f - iou + v + EPS_F);
    return 1.f - iou + d2 / c2 + alpha * v;
}

// ---------------- Kernel 1: binary cost matrix [B,Q,Q] (u8) ----------------
__global__ void cost_kernel(const float* __restrict__ curr, const float* __restrict__ prev,
                            const float* __restrict__ img, unsigned char* __restrict__ cost,
                            int B, int Q) {
    int g = blockIdx.x * blockDim.x + threadIdx.x;
    int QQ = Q * Q;
    if (g >= B * QQ) return;
    int b = g / QQ;
    int r = g - b * QQ;
    int i = r / Q;        // row: curr box index
    int j = r - i * Q;    // col: filtered (prev) box index
    float W = img[2 * b + 1], H = img[2 * b + 0];
    float4 cb = ld4(curr + (size_t)(b * Q + i) * 4);
    float4 pb = filt_adjust(ld4(prev + (size_t)(b * Q + j) * 4), W, H);
    float gl = giou_loss(to_xyxy(cb), to_xyxy(pb));
    cost[g] = (gl < GIOU_THR) ? (unsigned char)1 : (unsigned char)0;
}

// ---------------- Kernel 2: Hungarian (JV), one block per batch ----------------
// blockDim.x must equal n (=512). Exact replica of the reference's float64
// algorithm: with 0/1 costs all intermediates are exact small integers in f32.
__global__ void hungarian_kernel(const unsigned char* __restrict__ cost,
                                 int* __restrict__ cols, int n) {
    __shared__ float sh_u[513], sh_v[513], sh_minv[513];
    __shared__ int   sh_p[513], sh_way[513], sh_used[513];
    __shared__ float rv[512];
    __shared__ int   ri[512];
    __shared__ int   s_i0, s_j0, s_j1, s_done;
    __shared__ float s_delta;

    const int b   = blockIdx.x;
    const int tid = threadIdx.x;
    const unsigned char* cb = cost + (size_t)b * n * n;
    const float FINF = __builtin_inff();

    for (int j = tid; j <= n; j += blockDim.x) {
        sh_u[j] = 0.f; sh_v[j] = 0.f; sh_p[j] = 0; sh_way[j] = 0;
    }
    __syncthreads();

    for (int i = 1; i <= n; ++i) {
        if (tid == 0) { sh_p[0] = i; s_j0 = 0; }
        for (int j = tid; j <= n; j += blockDim.x) { sh_minv[j] = FINF; sh_used[j] = 0; }
        __syncthreads();

        for (;;) {
            // --- arrive at column j0 ---
            if (tid == 0) { sh_used[s_j0] = 1; s_i0 = sh_p[s_j0]; }
            __syncthreads();
            const int i0 = s_i0, j0 = s_j0;
            const float u_i0 = sh_u[i0];

            // --- scan row i0 over unused columns ---
            {
                int j = tid + 1;                       // j in 1..n
                if (!sh_used[j]) {
                    float cur = (float)cb[(size_t)(i0 - 1) * n + (j - 1)] - u_i0 - sh_v[j];
                    if (cur < sh_minv[j]) { sh_minv[j] = cur; sh_way[j] = j0; }
                }
            }
            __syncthreads();

            // --- argmin over unused columns (first-index tie-break, like np.argmin) ---
            {
                int j = tid + 1;
                rv[tid] = sh_used[j] ? FINF : sh_minv[j];
                ri[tid] = j;
            }
            __syncthreads();
            for (int s = 256; s > 0; s >>= 1) {
                if (tid < s) {
                    float ov = rv[tid + s]; int oi = ri[tid + s];
                    if (ov < rv[tid] || (ov == rv[tid] && oi < ri[tid])) {
                        rv[tid] = ov; ri[tid] = oi;
                    }
                }
                __syncthreads();
            }
            if (tid == 0) { s_j1 = ri[0]; s_delta = rv[0]; }
            __syncthreads();
            const float delta = s_delta;

            // --- dual update ---
            for (int j = tid; j <= n; j += blockDim.x) {
                if (sh_used[j]) {
                    sh_u[sh_p[j]] += delta;   // distinct p[j] per used column
                    sh_v[j]       -= delta;
                } else {
                    sh_minv[j]    -= delta;
                }
            }
            __syncthreads();
            if (tid == 0) { s_j0 = s_j1; s_done = (sh_p[s_j1] == 0); }
            __syncthreads();
            if (s_done) break;
        }

        // --- augment along alternating path ---
        if (tid == 0) {
            int jj = s_j0;
            while (jj != 0) { int jp = sh_way[jj]; sh_p[jj] = sh_p[jp]; jj = jp; }
        }
        __syncthreads();
    }

    // col_of_row[p[j]-1] = j-1
    {
        int j = tid + 1;
        cols[(size_t)b * n + (sh_p[j] - 1)] = j - 1;
    }
}

// ---------------- Kernel 3: matched CIoU mean; WMMA-based final reduction ----------------
// One block of 512 threads. Each thread sums 8 CIoU values; the 512 partials
// are reduced with 8 accumulating V_WMMA_F32_16X16X4_F32 ops (B = ones), then
// one xor-16 shuffle in wave32.
__global__ void ciou_kernel(const float* __restrict__ curr, const float* __restrict__ prev,
                            const float* __restrict__ img, const int* __restrict__ cols,
                            float* __restrict__ out, int B, int Q) {
    __shared__ float part[512];
    const int tid = threadIdx.x;
    const int BQ = B * Q;

    float s = 0.f;
    for (int g = tid; g < BQ; g += 512) {
        int b = g / Q;
        int q = g - b * Q;
        float W = img[2 * b + 1], H = img[2 * b + 0];
        int c = cols[(size_t)b * Q + q];
        float4 pboxc = ld4(curr + (size_t)(b * Q + c) * 4);          // matched pred
        float4 tbox  = filt_adjust(ld4(prev + (size_t)(b * Q + q) * 4), W, H);
        s += ciou_loss(to_xyxy(pboxc), to_xyxy(tbox));
    }
    part[tid] = s;
    __syncthreads();

    // WMMA reduction: every wave computes the full sum (uniform EXEC=all-ones).
    const int lane = tid & 31;
    const int m  = lane & 15;
    const int kb = (lane >> 4) * 2;                 // K pair: {0,1} or {2,3}
    v8f acc = {};
    v2f bones; bones[0] = 1.f; bones[1] = 1.f;
    for (int chunk = 0; chunk < 8; ++chunk) {
        const float* P = &part[chunk * 64];
        v2f a;
        a[0] = P[m * 4 + kb];
        a[1] = P[m * 4 + kb + 1];
        // D[m][n] = rowsum(A_m) + C[m][n]  (B all ones)
        acc = __builtin_amdgcn_wmma_f32_16x16x4_f32(
            /*neg_a=*/false, a, /*neg_b=*/false, bones,
            /*c_mod=*/(short)0, acc, /*reuse_a=*/false, /*reuse_b=*/false);
    }
    // lanes 0..15 hold rows 0..7 (one per VGPR), lanes 16..31 hold rows 8..15
    float rs = acc[0] + acc[1] + acc[2] + acc[3] + acc[4] + acc[5] + acc[6] + acc[7];
    float tot = rs + __shfl_xor(rs, 16, 32);
    if (tid == 0) out[0] = tot / (float)BQ;
}

extern "C" void kernel_launch(void* const* d_in, const int* in_sizes, int n_in,
                              void* d_out, int out_size, void* d_ws, size_t ws_size,
                              hipStream_t stream) {
    const float* prev_boxes  = (const float*)d_in[0];
    const float* curr_boxes  = (const float*)d_in[2];
    const float* image_sizes = (const float*)d_in[3];

    const int B = in_sizes[3] / 2;              // 8
    const int Q = in_sizes[0] / (4 * B);        // 512

    unsigned char* cost = (unsigned char*)d_ws;
    int* cols = (int*)((char*)d_ws + (size_t)B * Q * Q);

    const long total = (long)B * Q * Q;
    const int  blk   = 256;
    cost_kernel<<<(unsigned)((total + blk - 1) / blk), blk, 0, stream>>>(
        curr_boxes, prev_boxes, image_sizes, cost, B, Q);

    hungarian_kernel<<<B, Q, 0, stream>>>(cost, cols, Q);

    ciou_kernel<<<1, 512, 0, stream>>>(curr_boxes, prev_boxes, image_sizes, cols,
                                       (float*)d_out, B, Q);
}